// Track_81149112090741
// MI455X (gfx1250) — compile-verified
//
#include <hip/hip_runtime.h>
#include <stdint.h>

// CDNA5 TDM descriptor groups (CDNA5 ISA)
typedef unsigned int v4u __attribute__((ext_vector_type(4)));
typedef int          v8i __attribute__((ext_vector_type(8)));
typedef int          v4i __attribute__((ext_vector_type(4)));

#define TRK_CHUNK 1024  // steps staged in LDS per TDM flush (power of 2)

// Issue one TDM store of a contiguous LDS row -> contiguous global row.
// 1D tile: tile_dim0 = width elements of 4B, tensor_dim0 = tensor_w (>= width,
// so the OOB clip never triggers under either tile-local interpretation).
__device__ __forceinline__ void tdm_store_row(const float* gptr, unsigned lds_off,
                                              int width, int tensor_w) {
    uint64_t ga = (uint64_t)(uintptr_t)gptr;
    v4u g0;
    g0.x = 1u;                                   // count=1 (valid), user mode
    g0.y = lds_off;                              // lds_addr (byte offset)
    g0.z = (unsigned)(ga & 0xFFFFFFFFull);       // global_addr[31:0]
    g0.w = (unsigned)((ga >> 32) & 0x01FFFFFFu)  // global_addr[56:32]
         | (2u << 30);                           // type = 2 ("image")
    unsigned d0 = (unsigned)tensor_w;
    v8i g1;
    g1[0] = (int)(2u << 16);                     // wg_mask=0, data_size=2 (4B)
    g1[1] = (int)((d0 & 0xFFFFu) << 16);         // tensor_dim0[15:0]
    g1[2] = (int)((d0 >> 16) | (1u << 16));      // tensor_dim0[31:16], tensor_dim1=1
    g1[3] = (int)((unsigned)width << 16);        // tile_dim0 = width
    g1[4] = 1;                                   // tile_dim1=1, tile_dim2=0
    g1[5] = (int)d0;                             // tensor_dim0_stride[31:0] (benign)
    g1[6] = 0;                                   // stride0 hi, dim1_stride lo
    g1[7] = 0;                                   // dim1_stride hi
    v4i gz4 = {0, 0, 0, 0};                      // groups 2/3: dims/tiles unused
    v8i gz8 = {0, 0, 0, 0, 0, 0, 0, 0};          // trailing group (unused here)
    // This toolchain's 6-arg form: (g0, g1, g2, g3, extra, cpol)
    __builtin_amdgcn_tensor_store_from_lds(g0, g1, gz4, gz4, gz8, 0);
}

__global__ void __launch_bounds__(32, 1)
trk_rk4_serial(const float* __restrict__ timev,
               const float* __restrict__ r0v,
               const float* __restrict__ d0v,
               const float* __restrict__ gmp,
               const float* __restrict__ b0p,
               const float* __restrict__ kup,
               float* __restrict__ out,   // [6, n]: rows 0-2 = r, rows 3-5 = beta
               int n)
{
    __shared__ float lds[2][6][TRK_CHUNK];   // 48 KB double-buffered staging

    const float c     = 0.29979245f;
    const float invc  = 1.0f / c;
    const float invc2 = invc * invc;

    float dt  = timev[1] - timev[0];
    float dt2 = 0.5f * dt;
    float dt6 = dt * (1.0f / 6.0f);

    float gamma = gmp[0];
    float b0    = b0p[0];
    // v_cos_f32 computes cos(2*pi*x): fold 1/(2*pi) into k_u once.
    float k2    = kup[0] * 0.15915494309189535f;

    float d0x = d0v[0], d0y = d0v[1], d0z = d0v[2];
    float ps  = c * sqrtf(gamma * gamma - 1.0f) *
                __builtin_amdgcn_rsqf(d0x * d0x + d0y * d0y + d0z * d0z);
    float px = ps * d0x, py = ps * d0y, pz = ps * d0z;
    float rx = r0v[0], ry = r0v[1], rz = r0v[2];
    const float py2 = py * py;   // dp_y/dt == 0 exactly for B=(0,By,0)

    for (int i = 0; i < n; ++i) {
        int col = i & (TRK_CHUNK - 1);
        int buf = (i >> 10) & 1;
        if (col == 0) {
            // Before overwriting this LDS buffer, its store from 2 chunks ago
            // (6 row-DMAs) must have drained; at most the previous chunk's 6
            // may remain in flight (TDM ops complete in order per wave).
            __builtin_amdgcn_s_wait_tensorcnt(6);
        }

        // 1/gamma of current p — shared by beta output and RK4 stage 1
        float s1 = fmaf(px, px, fmaf(pz, pz, py2));
        float ig = __builtin_amdgcn_rsqf(fmaf(s1, invc2, 1.0f));
        float bf = ig * invc;                       // beta = p / (c*gamma)
        lds[buf][0][col] = rx;
        lds[buf][1][col] = ry;
        lds[buf][2][col] = rz;
        lds[buf][3][col] = px * bf;
        lds[buf][4][col] = py * bf;
        lds[buf][5][col] = pz * bf;

        bool last = (i == n - 1);
        if (col == (TRK_CHUNK - 1) || last) {
            int start = i - col;
            int width = col + 1;
            // LDS writes must be visible to the TDM engine before it reads.
            asm volatile("s_wait_dscnt 0" ::: "memory");
            #pragma unroll
            for (int r = 0; r < 6; ++r) {
                tdm_store_row(out + (size_t)r * n + start,
                              (unsigned)(uintptr_t)&lds[buf][r][0],
                              width, n);
            }
        }
        if (last) break;

        // ---- RK4 step (field only depends on z; p_y constant) ----
        // stage 1 (ig already computed)
        float by1 = b0 * __builtin_amdgcn_cosf(k2 * rz);
        float w1  = by1 * ig;
        float rk1x = px * ig, rk1y = py * ig, rk1z = pz * ig;
        float pk1x = pz * w1,  pk1z = -(px * w1);

        float pax = fmaf(pk1x, dt2, px);
        float paz = fmaf(pk1z, dt2, pz);
        float za  = fmaf(rk1z, dt2, rz);

        // stage 2
        float sa  = fmaf(pax, pax, fmaf(paz, paz, py2));
        float iga = __builtin_amdgcn_rsqf(fmaf(sa, invc2, 1.0f));
        float by2 = b0 * __builtin_amdgcn_cosf(k2 * za);
        float w2  = by2 * iga;
        float rk2x = pax * iga, rk2y = py * iga, rk2z = paz * iga;
        float pk2x = paz * w2,  pk2z = -(pax * w2);

        float pbx = fmaf(pk2x, dt2, px);
        float pbz = fmaf(pk2z, dt2, pz);
        float zb  = fmaf(rk2z, dt2, rz);

        // stage 3
        float sb  = fmaf(pbx, pbx, fmaf(pbz, pbz, py2));
        float igb = __builtin_amdgcn_rsqf(fmaf(sb, invc2, 1.0f));
        float by3 = b0 * __builtin_amdgcn_cosf(k2 * zb);
        float w3  = by3 * igb;
        float rk3x = pbx * igb, rk3y = py * igb, rk3z = pbz * igb;
        float pk3x = pbz * w3,  pk3z = -(pbx * w3);

        float pcx = fmaf(pk3x, dt, px);
        float pcz = fmaf(pk3z, dt, pz);
        float zc  = fmaf(rk3z, dt, rz);

        // stage 4
        float sc  = fmaf(pcx, pcx, fmaf(pcz, pcz, py2));
        float igc = __builtin_amdgcn_rsqf(fmaf(sc, invc2, 1.0f));
        float by4 = b0 * __builtin_amdgcn_cosf(k2 * zc);
        float w4  = by4 * igc;
        float rk4x = pcx * igc, rk4y = py * igc, rk4z = pcz * igc;
        float pk4x = pcz * w4,  pk4z = -(pcx * w4);

        // combine
        rx = fmaf(dt6, fmaf(2.0f, rk2x + rk3x, rk1x + rk4x), rx);
        ry = fmaf(dt6, fmaf(2.0f, rk2y + rk3y, rk1y + rk4y), ry);
        rz = fmaf(dt6, fmaf(2.0f, rk2z + rk3z, rk1z + rk4z), rz);
        px = fmaf(dt6, fmaf(2.0f, pk2x + pk3x, pk1x + pk4x), px);
        pz = fmaf(dt6, fmaf(2.0f, pk2z + pk3z, pk1z + pk4z), pz);
    }
    // S_ENDPGM implies wait-idle: all outstanding TDM stores drain here.
}

extern "C" void kernel_launch(void* const* d_in, const int* in_sizes, int n_in,
                              void* d_out, int out_size, void* d_ws, size_t ws_size,
                              hipStream_t stream) {
    (void)n_in; (void)out_size; (void)d_ws; (void)ws_size;
    const float* timev = (const float*)d_in[0];
    const float* r0    = (const float*)d_in[1];
    const float* d0    = (const float*)d_in[2];
    const float* gm    = (const float*)d_in[3];
    const float* b0    = (const float*)d_in[4];
    const float* ku    = (const float*)d_in[5];
    int n = in_sizes[0];
    // Single wave, single lane: the problem is one serial dependency chain.
    trk_rk4_serial<<<dim3(1), dim3(1), 0, stream>>>(timev, r0, d0, gm, b0, ku,
                                                    (float*)d_out, n);
}